// NonLocalBlock_81819126989596
// MI455X (gfx1250) — compile-verified
//
#include <hip/hip_runtime.h>

typedef float v2f __attribute__((ext_vector_type(2)));
typedef float v8f __attribute__((ext_vector_type(8)));

#define WMMA_F32(a, b, c) \
    __builtin_amdgcn_wmma_f32_16x16x4_f32(false, (a), false, (b), (short)0, (c), false, false)

// ---------------------------------------------------------------------------
// Kernel 1: per (batch, pooled row): A1 = W1@X+b1, A2 = W22@X+b22 over a
// 2-row (128-wide) spatial strip, then 2x2 maxpool -> g, phi (32 x 1024).
// 8 waves: waves 0-3 conv1 (g), waves 4-7 conv22 (phi); each wave does
// M=32 x N=32 of its conv (2 m-tiles x 2 n-tiles), K=64.
// ---------------------------------------------------------------------------
__global__ __launch_bounds__(256) void k1_conv_pool(
    const float* __restrict__ x,
    const float* __restrict__ w1, const float* __restrict__ b1,
    const float* __restrict__ w22, const float* __restrict__ b22,
    float* __restrict__ g, float* __restrict__ phi)
{
    const int blk = blockIdx.x;       // 16 batches * 32 pooled rows
    const int b  = blk >> 5;
    const int pr = blk & 31;
    const float* Xb = x + (size_t)b * 64 * 4096;
    const int sbase = pr * 128;       // two spatial rows, contiguous

    const int t    = threadIdx.x;
    const int wv   = t >> 5;
    const int lane = t & 31;
    const int conv = wv >> 2;         // 0: conv1, 1: conv22
    const int w4   = wv & 3;          // n-pair index within conv
    const float* W    = conv ? w22 : w1;
    const float* bias = conv ? b22 : b1;

    __shared__ float At[2][32][128];  // 32 KB

    const int l15 = lane & 15;
    const int klo = (lane >> 4) << 1; // 0 or 2
    const int n0  = w4 * 32;

    v8f acc[4] = {};                  // [mi*2 + nj]
    #pragma unroll 4
    for (int k = 0; k < 64; k += 4) {
        v2f a0, a1, bf0, bf1;
        const float* pa0 = W + (l15)      * 64 + k + klo;
        const float* pa1 = W + (16 + l15) * 64 + k + klo;
        a0.x = pa0[0]; a0.y = pa0[1];
        a1.x = pa1[0]; a1.y = pa1[1];
        const float* pb0 = Xb + (size_t)(k + klo) * 4096 + sbase + n0 + l15;
        bf0.x = pb0[0];  bf0.y = pb0[4096];
        bf1.x = pb0[16]; bf1.y = pb0[16 + 4096];
        acc[0] = WMMA_F32(a0, bf0, acc[0]);
        acc[1] = WMMA_F32(a0, bf1, acc[1]);
        acc[2] = WMMA_F32(a1, bf0, acc[2]);
        acc[3] = WMMA_F32(a1, bf1, acc[3]);
    }

    const int rowadd = (lane >> 4) * 8;
    #pragma unroll
    for (int mi = 0; mi < 2; ++mi)
        #pragma unroll
        for (int nj = 0; nj < 2; ++nj) {
            v8f v = acc[mi * 2 + nj];
            #pragma unroll
            for (int j = 0; j < 8; ++j) {
                const int c = mi * 16 + j + rowadd;
                At[conv][c][n0 + nj * 16 + l15] = v[j] + bias[c];
            }
        }
    __syncthreads();

    // 2x2 maxpool: rows are local n 0..63 and 64..127
    const int sel = t >> 7;           // 0 -> g, 1 -> phi
    const int t2  = t & 127;
    const int ch  = t2 >> 2;
    const int wcb = (t2 & 3) * 8;
    float* outp = sel ? phi : g;
    #pragma unroll
    for (int wc = wcb; wc < wcb + 8; ++wc) {
        float m = fmaxf(fmaxf(At[sel][ch][2 * wc],      At[sel][ch][2 * wc + 1]),
                        fmaxf(At[sel][ch][64 + 2 * wc], At[sel][ch][64 + 2 * wc + 1]));
        outp[(size_t)b * 32768 + (size_t)ch * 1024 + pr * 32 + wc] = m;
    }
}

// ---------------------------------------------------------------------------
// Kernel 2 (1 block / batch, 128 thr): kv = phi @ g^T / 1024 (WMMA, K=1024),
// then T = kv^T@W21, P = W3@T, q = W3@(kv^T@b21) + b3 (tiny, plain VALU).
// ---------------------------------------------------------------------------
__global__ __launch_bounds__(128) void k2_kv_P(
    const float* __restrict__ g, const float* __restrict__ phi,
    const float* __restrict__ w21, const float* __restrict__ b21,
    const float* __restrict__ w3,  const float* __restrict__ b3,
    float* __restrict__ Pg, float* __restrict__ qg)
{
    const int b = blockIdx.x;
    const int t = threadIdx.x, wv = t >> 5, lane = t & 31;
    const float* gb = g   + (size_t)b * 32768;
    const float* pb = phi + (size_t)b * 32768;

    __shared__ float kv_s[32][32];
    __shared__ float T_s[32][64];
    __shared__ float tb_s[32];

    const int m0 = (wv >> 1) * 16, n0 = (wv & 1) * 16;
    const int l15 = lane & 15;
    const int klo = (lane >> 4) << 1;

    v8f acc = {};
    #pragma unroll 8
    for (int k = 0; k < 1024; k += 4) {
        v2f a, bf;
        const float* pa = pb + (size_t)(m0 + l15) * 1024 + k + klo;
        a.x = pa[0]; a.y = pa[1];
        const float* pg = gb + (size_t)(n0 + l15) * 1024 + k + klo;
        bf.x = pg[0]; bf.y = pg[1];
        acc = WMMA_F32(a, bf, acc);
    }
    const int rowadd = (lane >> 4) * 8;
    #pragma unroll
    for (int j = 0; j < 8; ++j)
        kv_s[m0 + j + rowadd][n0 + l15] = acc[j] * (1.0f / 1024.0f);
    __syncthreads();

    // T[j][c] = sum_i kv[i][j] * w21[i][c]
    for (int idx = t; idx < 2048; idx += 128) {
        const int j = idx >> 6, c = idx & 63;
        float s = 0.f;
        #pragma unroll
        for (int i = 0; i < 32; ++i) s += kv_s[i][j] * w21[i * 64 + c];
        T_s[j][c] = s;
    }
    if (t < 32) {
        float s = 0.f;
        #pragma unroll
        for (int i = 0; i < 32; ++i) s += kv_s[i][t] * b21[i];
        tb_s[t] = s;
    }
    __syncthreads();

    // P[o][c] = sum_j w3[o][j] * T[j][c] ; q[o] = sum_j w3[o][j]*tb[j] + b3[o]
    for (int idx = t; idx < 4096; idx += 128) {
        const int o = idx >> 6, c = idx & 63;
        float s = 0.f;
        #pragma unroll
        for (int j = 0; j < 32; ++j) s += w3[o * 32 + j] * T_s[j][c];
        Pg[(size_t)b * 4096 + o * 64 + c] = s;
    }
    if (t < 64) {
        float s = b3[t];
        #pragma unroll
        for (int j = 0; j < 32; ++j) s += w3[t * 32 + j] * tb_s[j];
        qg[b * 64 + t] = s;
    }
}

// ---------------------------------------------------------------------------
// Shared WMMA GEMM: z(64 x 256) = P_b(64x64) @ X_slice(64x256), per wave:
// all 4 m-tiles x 2 n-tiles (n0 = wv*32), K=64.
// ---------------------------------------------------------------------------
__device__ __forceinline__ void gemm_z(const float* __restrict__ Pb,
                                       const float* __restrict__ Xb,
                                       int s0, int lane, int wv, v8f acc[8])
{
    const int l15 = lane & 15;
    const int klo = (lane >> 4) << 1;
    __builtin_prefetch(Xb + s0 + wv * 32 + l15, 0, 1);   // global_prefetch_b8
    #pragma unroll 4
    for (int k = 0; k < 64; k += 4) {
        v2f a[4], bf[2];
        #pragma unroll
        for (int mi = 0; mi < 4; ++mi) {
            const float* pa = Pb + (mi * 16 + l15) * 64 + k + klo;
            a[mi].x = pa[0]; a[mi].y = pa[1];
        }
        #pragma unroll
        for (int nj = 0; nj < 2; ++nj) {
            const float* pb = Xb + (size_t)(k + klo) * 4096 + s0 + wv * 32 + nj * 16 + l15;
            bf[nj].x = pb[0]; bf[nj].y = pb[4096];
        }
        #pragma unroll
        for (int mi = 0; mi < 4; ++mi)
            #pragma unroll
            for (int nj = 0; nj < 2; ++nj)
                acc[mi * 2 + nj] = WMMA_F32(a[mi], bf[nj], acc[mi * 2 + nj]);
    }
}

// ---------------------------------------------------------------------------
// Kernel 3: stats pass. z = P_b@X + q; per-channel sum & sumsq via
// half-wave shuffle reduction -> LDS atomics -> global atomics.
// ---------------------------------------------------------------------------
__global__ __launch_bounds__(256) void k3_stats(
    const float* __restrict__ x, const float* __restrict__ Pg,
    const float* __restrict__ qg, float* __restrict__ gsum, float* __restrict__ gsq)
{
    const int blk = blockIdx.x;      // 16 batches * 16 chunks
    const int b = blk >> 4, ci = blk & 15;
    const int s0 = ci * 256;
    const float* Xb = x  + (size_t)b * 262144;
    const float* Pb = Pg + (size_t)b * 4096;
    const int t = threadIdx.x, wv = t >> 5, lane = t & 31;

    __shared__ float qs[64], csum[64], csq[64];
    if (t < 64) { qs[t] = qg[b * 64 + t]; csum[t] = 0.f; csq[t] = 0.f; }
    __syncthreads();

    v8f acc[8] = {};
    gemm_z(Pb, Xb, s0, lane, wv, acc);

    const int rowadd = (lane >> 4) * 8;
    #pragma unroll
    for (int mi = 0; mi < 4; ++mi)
        #pragma unroll
        for (int nj = 0; nj < 2; ++nj) {
            v8f v = acc[mi * 2 + nj];
            #pragma unroll
            for (int j = 0; j < 8; ++j) {
                const int c = mi * 16 + j + rowadd;
                float z = v[j] + qs[c];
                float s = z, sq = z * z;
                // reduce over the 16 lanes that share this channel
                #pragma unroll
                for (int m = 1; m <= 8; m <<= 1) {
                    s  += __shfl_xor(s,  m, 32);
                    sq += __shfl_xor(sq, m, 32);
                }
                if ((lane & 15) == 0) {
                    atomicAdd(&csum[c], s);
                    atomicAdd(&csq[c], sq);
                }
            }
        }
    __syncthreads();
    if (t < 64) { atomicAdd(&gsum[t], csum[t]); atomicAdd(&gsq[t], csq[t]); }
}

// ---------------------------------------------------------------------------
// Kernel 4: fold batch stats into per-channel scale/shift.
// ---------------------------------------------------------------------------
__global__ void k4_finalize(const float* __restrict__ gsum, const float* __restrict__ gsq,
                            const float* __restrict__ gamma, const float* __restrict__ beta,
                            float* __restrict__ scale, float* __restrict__ shift)
{
    const int c = threadIdx.x;
    if (c < 64) {
        const float inv_n = 1.0f / 65536.0f;   // B*H*W = 16*64*64
        float mu  = gsum[c] * inv_n;
        float var = gsq[c] * inv_n - mu * mu;
        float sc  = gamma[c] * rsqrtf(var + 1e-5f);
        scale[c] = sc;
        shift[c] = beta[c] - mu * sc;
    }
}

// ---------------------------------------------------------------------------
// Kernel 5: recompute z, apply BN + residual, write output from fragments.
// ---------------------------------------------------------------------------
__global__ __launch_bounds__(256) void k5_out(
    const float* __restrict__ x, const float* __restrict__ Pg,
    const float* __restrict__ qg, const float* __restrict__ scale,
    const float* __restrict__ shift, float* __restrict__ out)
{
    const int blk = blockIdx.x;
    const int b = blk >> 4, ci = blk & 15;
    const int s0 = ci * 256;
    const float* Xb = x  + (size_t)b * 262144;
    const float* Pb = Pg + (size_t)b * 4096;
    const int t = threadIdx.x, wv = t >> 5, lane = t & 31;

    __shared__ float qs[64], scs[64], shs[64];
    if (t < 64) { qs[t] = qg[b * 64 + t]; scs[t] = scale[t]; shs[t] = shift[t]; }
    __syncthreads();

    v8f acc[8] = {};
    gemm_z(Pb, Xb, s0, lane, wv, acc);

    const int rowadd = (lane >> 4) * 8;
    const int l15 = lane & 15;
    float* outb = out + (size_t)b * 262144;
    #pragma unroll
    for (int mi = 0; mi < 4; ++mi)
        #pragma unroll
        for (int nj = 0; nj < 2; ++nj) {
            v8f v = acc[mi * 2 + nj];
            const int n = s0 + wv * 32 + nj * 16 + l15;
            #pragma unroll
            for (int j = 0; j < 8; ++j) {
                const int c = mi * 16 + j + rowadd;
                const size_t idx = (size_t)c * 4096 + n;
                float z = v[j] + qs[c];
                outb[idx] = scs[c] * z + shs[c] + Xb[idx];
            }
        }
}

// ---------------------------------------------------------------------------
extern "C" void kernel_launch(void* const* d_in, const int* in_sizes, int n_in,
                              void* d_out, int out_size, void* d_ws, size_t ws_size,
                              hipStream_t stream)
{
    const float* x     = (const float*)d_in[0];
    const float* w1    = (const float*)d_in[1];
    const float* b1    = (const float*)d_in[2];
    const float* w21   = (const float*)d_in[3];
    const float* b21   = (const float*)d_in[4];
    const float* w22   = (const float*)d_in[5];
    const float* b22   = (const float*)d_in[6];
    const float* w3    = (const float*)d_in[7];
    const float* b3    = (const float*)d_in[8];
    const float* gamma = (const float*)d_in[9];
    const float* beta  = (const float*)d_in[10];

    float* ws    = (float*)d_ws;
    float* g     = ws;                 // 16*32*1024
    float* phi   = ws + 524288;        // 16*32*1024
    float* Pg    = ws + 1048576;       // 16*64*64
    float* qg    = ws + 1114112;       // 16*64
    float* gsum  = ws + 1115136;       // 64
    float* gsq   = gsum + 64;          // 64
    float* scale = gsum + 128;         // 64
    float* shift = gsum + 192;         // 64

    hipMemsetAsync(gsum, 0, 128 * sizeof(float), stream);

    k1_conv_pool<<<512, 256, 0, stream>>>(x, w1, b1, w22, b22, g, phi);
    k2_kv_P<<<16, 128, 0, stream>>>(g, phi, w21, b21, w3, b3, Pg, qg);
    k3_stats<<<256, 256, 0, stream>>>(x, Pg, qg, gsum, gsq);
    k4_finalize<<<1, 64, 0, stream>>>(gsum, gsq, gamma, beta, scale, shift);
    k5_out<<<256, 256, 0, stream>>>(x, Pg, qg, scale, shift, (float*)d_out);
}